// GraphAttentionLayer_90907277787254
// MI455X (gfx1250) — compile-verified
//
#include <hip/hip_runtime.h>

#define BATCH 16
#define NNODE 2048
#define FDIM  64
#define LDS_STRIDE 516   // 512 + 4: keeps float4 alignment, spreads LDS banks (516 % 64 == 4)

typedef float v2f __attribute__((ext_vector_type(2)));
typedef float v8f __attribute__((ext_vector_type(8)));

// ---------------------------------------------------------------------------
// Kernel 1: h[b,n,o] = sum_f X[b,n,f] * W[o,f]   (X @ W^T) via WMMA f32 16x16x4
// One wave (32 threads) computes one 16(M=nodes) x 16(N=out-feat) tile, K=64.
// A layout (16x4 f32): lanes 0-15 hold M=0..15; v0 = K[0|2], v1 = K[1|3]
// B layout (4x16 f32): lanes 0-15 -> N=0..15;   v0 = K[0|2], v1 = K[1|3]
// ---------------------------------------------------------------------------
__global__ __launch_bounds__(32) void gat_h_kernel(const float* __restrict__ X,
                                                   const float* __restrict__ W,
                                                   float* __restrict__ h) {
  const int bo   = blockIdx.x;
  const int ot   = bo & 3;            // which 16-wide slice of F_OUT
  const int it   = (bo >> 2) & 127;   // which 16-row slice of N
  const int b    = bo >> 9;
  const int i0   = it << 4;
  const int o0   = ot << 4;
  const int lane = threadIdx.x;
  const int mrow = lane & 15;         // M row (for A) / N col (for B)
  const int hi   = lane >> 4;         // selects K pair {0,1} vs {2,3}

  const float* Xp = X + ((size_t)(b * NNODE + i0 + mrow)) * FDIM + (hi << 1);
  const float* Wp = W + ((size_t)(o0 + mrow)) * FDIM + (hi << 1);

  v8f acc = {0.f, 0.f, 0.f, 0.f, 0.f, 0.f, 0.f, 0.f};
#pragma unroll
  for (int f0 = 0; f0 < FDIM; f0 += 4) {
    const float2 av = *reinterpret_cast<const float2*>(Xp + f0);
    const float2 bv = *reinterpret_cast<const float2*>(Wp + f0);
    v2f A = {av.x, av.y};
    v2f Bm = {bv.x, bv.y};
    acc = __builtin_amdgcn_wmma_f32_16x16x4_f32(false, A, false, Bm,
                                                (short)0, acc, false, false);
  }
  // D layout: lane 0-15 -> M=v, lane 16-31 -> M=v+8; N = lane&15
  float* hp = h + ((size_t)(b * NNODE + i0 + (hi << 3))) * FDIM + o0 + mrow;
#pragma unroll
  for (int v = 0; v < 8; ++v)
    hp[(size_t)v * FDIM] = acc[v];
}

// ---------------------------------------------------------------------------
// Kernel 2: s_i = h @ a_src, s_j = h @ a_dst  (one thread per (b,n))
// ---------------------------------------------------------------------------
__global__ __launch_bounds__(256) void gat_s_kernel(const float* __restrict__ h,
                                                    const float* __restrict__ a,
                                                    float* __restrict__ s_src,
                                                    float* __restrict__ s_dst) {
  const int idx = blockIdx.x * 256 + threadIdx.x;   // == b*N + n
  const float* hr = h + (size_t)idx * FDIM;
  float ss = 0.f, sd = 0.f;
#pragma unroll 8
  for (int f = 0; f < FDIM; ++f) {
    const float hv = hr[f];
    ss += hv * a[f];
    sd += hv * a[FDIM + f];
  }
  s_src[idx] = ss;
  s_dst[idx] = sd;
}

// score recompute helper (mask/adj_w live in L2: 2x16MB reused by all batches)
__device__ __forceinline__ float4 score4(const float* __restrict__ mrow,
                                         const float* __restrict__ wrow,
                                         const float* __restrict__ sjb,
                                         float si, int j) {
  const float4 mk = *reinterpret_cast<const float4*>(mrow + j);
  const float4 aw = *reinterpret_cast<const float4*>(wrow + j);
  const float4 sj = *reinterpret_cast<const float4*>(sjb + j);
  float4 r;
  r.x = (si + sj.x) * mk.x * (1.0f / (1.0f + __expf(-aw.x)));
  r.y = (si + sj.y) * mk.y * (1.0f / (1.0f + __expf(-aw.y)));
  r.z = (si + sj.z) * mk.z * (1.0f / (1.0f + __expf(-aw.z)));
  r.w = (si + sj.w) * mk.w * (1.0f / (1.0f + __expf(-aw.w)));
  return r;
}

// ---------------------------------------------------------------------------
// Kernel 3: fused scores -> softmax -> attn (written to HBM) -> attn @ h.
// One workgroup (256 thr / 8 waves) per (b, 16-row tile of i).
// Streaming softmax: pass1 row-max, pass2 row-sum, pass3 per-512-col chunk:
// materialize attn chunk in LDS + HBM, then split-K WMMA (4 O-tiles x 2 K-halves).
// ---------------------------------------------------------------------------
__global__ __launch_bounds__(256) void gat_attn_kernel(
    const float* __restrict__ mask, const float* __restrict__ adjw,
    const float* __restrict__ h, const float* __restrict__ s_src,
    const float* __restrict__ s_dst, float* __restrict__ updated,
    float* __restrict__ attn) {
  __shared__ float sc[16 * LDS_STRIDE];   // 32.25 KB attn chunk (16 x 512, stride 516)
  __shared__ float red[16][16];           // per-row partial reductions
  __shared__ float rowstat[16];           // row max, then row sum
  __shared__ float part[4][8][32];        // split-K partial accumulators

  const int b   = blockIdx.x >> 7;
  const int it  = blockIdx.x & 127;
  const int i0  = it << 4;
  const int t   = threadIdx.x;
  const int r   = t >> 4;                 // row within tile (0..15)
  const int c   = t & 15;                 // 16 threads per row
  const int row = i0 + r;

  const float  si   = s_src[b * NNODE + row];
  const float* mrow = mask + (size_t)row * NNODE;
  const float* wrow = adjw + (size_t)row * NNODE;
  const float* sjb  = s_dst + b * NNODE;

  // ---- pass 1: row max (scores over all 2048 j, incl. masked zeros) ----
  float mx = -3.0e38f;
  for (int s = 0; s < 32; ++s) {
    const float4 v = score4(mrow, wrow, sjb, si, (c << 2) + (s << 6));
    mx = fmaxf(mx, fmaxf(fmaxf(v.x, v.y), fmaxf(v.z, v.w)));
  }
  red[r][c] = mx;
  __syncthreads();
  if (t < 16) {
    float m2 = red[t][0];
#pragma unroll
    for (int q = 1; q < 16; ++q) m2 = fmaxf(m2, red[t][q]);
    rowstat[t] = m2;
  }
  __syncthreads();
  mx = rowstat[r];

  // ---- pass 2: row sum of exp(score - max) ----
  float sm = 0.f;
  for (int s = 0; s < 32; ++s) {
    const float4 v = score4(mrow, wrow, sjb, si, (c << 2) + (s << 6));
    sm += __expf(v.x - mx) + __expf(v.y - mx) + __expf(v.z - mx) + __expf(v.w - mx);
  }
  red[r][c] = sm;
  __syncthreads();
  if (t < 16) {
    float s2 = red[t][0];
#pragma unroll
    for (int q = 1; q < 16; ++q) s2 += red[t][q];
    rowstat[t] = s2;
  }
  __syncthreads();
  const float inv = 1.0f / rowstat[r];

  // WMMA work split: wave -> (O-tile, K-half)
  const int wave = t >> 5, lane = t & 31;
  const int ot = wave & 3, kh = wave >> 2;
  const int o0 = ot << 4;
  const int ar = lane & 15;               // A row / B col
  const int hi = lane >> 4;               // K-pair selector
  v8f acc = {0.f, 0.f, 0.f, 0.f, 0.f, 0.f, 0.f, 0.f};

  // ---- pass 3: 4 chunks of 512 columns ----
  for (int ch = 0; ch < 4; ++ch) {
    const int j0 = ch << 9;
    // normalized attn for this chunk -> LDS + HBM (coalesced float4)
    for (int s = 0; s < 8; ++s) {
      const int jl = (c << 2) + (s << 6);
      const float4 v = score4(mrow, wrow, sjb, si, j0 + jl);
      float4 av;
      av.x = __expf(v.x - mx) * inv;
      av.y = __expf(v.y - mx) * inv;
      av.z = __expf(v.z - mx) * inv;
      av.w = __expf(v.w - mx) * inv;
      *reinterpret_cast<float4*>(&sc[r * LDS_STRIDE + jl]) = av;
      *reinterpret_cast<float4*>(attn + ((size_t)(b * NNODE + row)) * NNODE + j0 + jl) = av;
    }
    __syncthreads();
    // 16x16 tile += attn_chunk[16 x 256(K-half)] @ h[256 x 16]
    const float* hbase =
        h + ((size_t)(b * NNODE + j0 + (kh << 8) + (hi << 1))) * FDIM + o0 + ar;
#pragma unroll 4
    for (int kk = 0; kk < 256; kk += 4) {
      const int k = (kh << 8) + kk;
      const float2 av2 =
          *reinterpret_cast<const float2*>(&sc[ar * LDS_STRIDE + k + (hi << 1)]);
      v2f A = {av2.x, av2.y};
      const float* hp = hbase + (size_t)kk * FDIM;
      v2f Bm = {hp[0], hp[FDIM]};
      acc = __builtin_amdgcn_wmma_f32_16x16x4_f32(false, A, false, Bm,
                                                  (short)0, acc, false, false);
    }
    __syncthreads();
  }

  // ---- split-K reduction + store updated ----
  if (kh == 1) {
#pragma unroll
    for (int v = 0; v < 8; ++v) part[ot][v][lane] = acc[v];
  }
  __syncthreads();
  if (kh == 0) {
    float* up = updated + ((size_t)(b * NNODE + i0 + (hi << 3))) * FDIM + o0 + ar;
#pragma unroll
    for (int v = 0; v < 8; ++v)
      up[(size_t)v * FDIM] = acc[v] + part[ot][v][lane];
  }
}

extern "C" void kernel_launch(void* const* d_in, const int* in_sizes, int n_in,
                              void* d_out, int out_size, void* d_ws, size_t ws_size,
                              hipStream_t stream) {
  const float* X    = (const float*)d_in[0];   // node_features (B,N,F_IN)
  const float* mask = (const float*)d_in[1];   // adjacency_mask (N,N)
  const float* W    = (const float*)d_in[2];   // W (F_OUT,F_IN)
  const float* a    = (const float*)d_in[3];   // a (1, 2*F_OUT)
  const float* adjw = (const float*)d_in[4];   // adj_weights (N,N)

  float* updated = (float*)d_out;                                   // (B,N,F_OUT)
  float* attn    = updated + (size_t)BATCH * NNODE * FDIM;          // (B,N,N)

  float* h     = (float*)d_ws;                                      // (B,N,F_OUT)
  float* s_src = h + (size_t)BATCH * NNODE * FDIM;
  float* s_dst = s_src + BATCH * NNODE;

  gat_h_kernel<<<BATCH * (NNODE / 16) * (FDIM / 16), 32, 0, stream>>>(X, W, h);
  gat_s_kernel<<<(BATCH * NNODE) / 256, 256, 0, stream>>>(h, a, s_src, s_dst);
  gat_attn_kernel<<<BATCH * (NNODE / 16), 256, 0, stream>>>(mask, adjw, h, s_src,
                                                            s_dst, updated, attn);
}